// Attention_7593502179526
// MI455X (gfx1250) — compile-verified
//
#include <hip/hip_runtime.h>
#include <math.h>

// Problem constants (B, S1, S2, DV) = (64, 1024, 1024, 256)
#define BATCH 64
#define S1 1024
#define S2 1024
#define DV 256
#define JCHUNKS 8                 // j-split per batch for the weighted sum
#define JPC (S2 / JCHUNKS)        // 128 j's per chunk

typedef __attribute__((ext_vector_type(2))) float v2f;
typedef __attribute__((ext_vector_type(8))) float v8f;

// ---------------------------------------------------------------------------
// Kernel 1: per-batch softmax weights p[b,j] = softmax_j(key[b,j,:]·w_k)
// (sx and bias cancel exactly under softmax over j)
// grid = (BATCH), block = 256
// ---------------------------------------------------------------------------
__global__ void softmax_p_kernel(const float* __restrict__ key,
                                 const float* __restrict__ W,
                                 float* __restrict__ p) {
    const int b   = blockIdx.x;
    const int tid = threadIdx.x;
    __shared__ float red[256];

    const float wk0 = W[3], wk1 = W[4], wk2 = W[5];
    const float* kb = key + (size_t)b * S2 * 3;

    float sk[4];
    float m = -INFINITY;
#pragma unroll
    for (int t = 0; t < 4; ++t) {
        const int j = tid + t * 256;
        const float* kp = kb + j * 3;
        float s = fmaf(kp[0], wk0, fmaf(kp[1], wk1, kp[2] * wk2));
        sk[t] = s;
        m = fmaxf(m, s);
    }
    red[tid] = m;
    __syncthreads();
    for (int off = 128; off > 0; off >>= 1) {
        if (tid < off) red[tid] = fmaxf(red[tid], red[tid + off]);
        __syncthreads();
    }
    const float gmax = red[0];
    __syncthreads();

    float e[4];
    float lsum = 0.f;
#pragma unroll
    for (int t = 0; t < 4; ++t) {
        e[t] = expf(sk[t] - gmax);
        lsum += e[t];
    }
    red[tid] = lsum;
    __syncthreads();
    for (int off = 128; off > 0; off >>= 1) {
        if (tid < off) red[tid] += red[tid + off];
        __syncthreads();
    }
    const float inv = 1.0f / red[0];
#pragma unroll
    for (int t = 0; t < 4; ++t)
        p[(size_t)b * S2 + tid + t * 256] = e[t] * inv;
}

// ---------------------------------------------------------------------------
// Kernel 2: partial weighted sums via V_WMMA_F32_16X16X4_F32 (exact fp32 FMA).
//   part[b,chunk,d] = sum_{j in chunk} p[b,j] * value[b,j,d]
// grid = (JCHUNKS, BATCH), block = 256 (8 waves; each wave owns 2 d-chunks).
// A(16x4) = V^T tile (lane<16: K=0,1 ; lane>=16: K=2,3 per ISA layout)
// B(4x16) = p[j0..j0+3] broadcast over N -> every D column equals c-partial.
// ---------------------------------------------------------------------------
__global__ void wsum_kernel(const float* __restrict__ value,
                            const float* __restrict__ p,
                            float* __restrict__ part) {
    const int b     = blockIdx.y;
    const int chunk = blockIdx.x;
    const int wave  = threadIdx.x >> 5;
    const int lane  = threadIdx.x & 31;
    const int half  = lane >> 4;   // 0: K=0,1   1: K=2,3
    const int col   = lane & 15;

    const float* Vb = value + (size_t)b * S2 * DV;
    const float* pb = p + (size_t)b * S2;
    const int jbase = chunk * JPC;
    float* outp = part + ((size_t)b * JCHUNKS + chunk) * DV;

#pragma unroll
    for (int c = 0; c < 2; ++c) {
        const int dbase = (wave * 2 + c) * 16;
        v8f acc = {};
        for (int js = 0; js < JPC; js += 4) {
            const int j = jbase + js + 2 * half;
            v2f A, Bm;
            A.x  = Vb[(size_t)j * DV + dbase + col];        // A[M=col, K=2h+0]
            A.y  = Vb[(size_t)(j + 1) * DV + dbase + col];  // A[M=col, K=2h+1]
            Bm.x = pb[j];                                   // B[K=2h+0, N=*]
            Bm.y = pb[j + 1];                               // B[K=2h+1, N=*]
            acc = __builtin_amdgcn_wmma_f32_16x16x4_f32(
                /*neg_a=*/false, A, /*neg_b=*/false, Bm,
                /*c_mod=*/(short)0, acc, /*reuse_a=*/false, /*reuse_b=*/false);
        }
        // D layout: VGPR r -> M=r (lanes 0-15) / M=8+r (lanes 16-31); N = col.
        if (col == 0) {
#pragma unroll
            for (int r = 0; r < 8; ++r)
                outp[dbase + 8 * half + r] = acc[r];
        }
    }
}

// ---------------------------------------------------------------------------
// Kernel 3: reduce the JCHUNKS partials and broadcast over i (rows of out).
// grid = (BATCH * S1/16), block = 256; each thread owns column d = tid and
// writes 16 rows — fully coalesced 1KB row stores, saturates HBM write BW.
// ---------------------------------------------------------------------------
__global__ void bcast_kernel(const float* __restrict__ part,
                             float* __restrict__ out) {
    const int blk    = blockIdx.x;
    const int b      = blk >> 6;        // / (S1/16)
    const int rowblk = blk & 63;
    const int tid    = threadIdx.x;     // d column

    const float* pp = part + (size_t)b * JCHUNKS * DV + tid;
    float s = 0.f;
#pragma unroll
    for (int k = 0; k < JCHUNKS; ++k) s += pp[k * DV];

    float* ob = out + ((size_t)b * S1 + rowblk * 16) * DV + tid;
#pragma unroll
    for (int r = 0; r < 16; ++r) ob[(size_t)r * DV] = s;
}

// ---------------------------------------------------------------------------
extern "C" void kernel_launch(void* const* d_in, const int* in_sizes, int n_in,
                              void* d_out, int out_size, void* d_ws, size_t ws_size,
                              hipStream_t stream) {
    (void)in_sizes; (void)n_in; (void)out_size; (void)ws_size;
    // setup_inputs order: x, key, value, W, b
    // x (d_in[0]) and b (d_in[4]) cancel exactly under softmax over j.
    const float* key   = (const float*)d_in[1];
    const float* value = (const float*)d_in[2];
    const float* W     = (const float*)d_in[3];
    float* out = (float*)d_out;

    float* p    = (float*)d_ws;                              // BATCH*S2 floats (256 KB)
    float* part = p + (size_t)BATCH * S2;                    // BATCH*JCHUNKS*DV floats (512 KB)

    softmax_p_kernel<<<dim3(BATCH), dim3(256), 0, stream>>>(key, W, p);
    wsum_kernel<<<dim3(JCHUNKS, BATCH), dim3(256), 0, stream>>>(value, p, part);
    bcast_kernel<<<dim3(BATCH * (S1 / 16)), dim3(256), 0, stream>>>(part, out);
}